// CDAM_67714454389366
// MI455X (gfx1250) — compile-verified
//
#include <hip/hip_runtime.h>
#include <hip/hip_bf16.h>

// ---------------------------------------------------------------------------
// MI455X (gfx1250) implementation.
//
// Phase 1 (tiny, 1 wave32): q/k projections via V_WMMA_F32_16X16X4_F32
//   (native fp32 WMMA -> bit-compatible with the fp32 reference math),
//   scores, softmax. Writes attn_flat to d_out tail, softmax weights to d_ws.
// Phase 2 (HBM-bound, ~320 MB @ 23.3 TB/s ~= 14 us): res = sum_n a[b,n]*v[b,n]
//   streamed with non-temporal B128 loads/stores (v = 256 MB > 192 MB L2,
//   touched once -> NT avoids cache pollution).
// ---------------------------------------------------------------------------

typedef float v2f __attribute__((ext_vector_type(2)));
typedef float v4f __attribute__((ext_vector_type(4)));
typedef float v8f __attribute__((ext_vector_type(8)));

__device__ __forceinline__ v8f wmma_f32_k4(v2f a, v2f b, v8f c) {
    // 8 args: (neg_a, A, neg_b, B, c_mod, C, reuse_a, reuse_b)
    return __builtin_amdgcn_wmma_f32_16x16x4_f32(
        false, a, false, b, (short)0, c, false, false);
}

// ---------------------------------------------------------------------------
// Phase 1: projections + scores + softmax.  Launch: <<<1, 32>>> (one wave32,
// EXEC all-ones as required by WMMA).
//
//   qh = Qpad(16x64) @ Wq^T(64x64)   (rows 0..3 valid, rest zero-padded)
//   kh = K(16x64)    @ Wk^T(64x64)   (rows = (b,n) flattened)
//   attn[r=4b+n] = (qh[b] . kh[r]) / sqrt(64)
//   weights[b][n] = softmax_n(attn[b][:])
//
// V_WMMA_F32_16X16X4_F32 VGPR layouts (ISA 7.12.2):
//   A (16x4):  lane = m + 16*(k>=2); vgpr pair holds (k, k+1) for that half
//   B (4x16):  lane l<16 -> col l, rows (0,1); lane>=16 -> col l-16, rows (2,3)
//   C/D (16x16): vgpr p, lanes 0-15 -> row p, lanes 16-31 -> row p+8
// ---------------------------------------------------------------------------
__global__ __launch_bounds__(32)
void CDAM_attn_wmma_kernel(const float* __restrict__ q,    // 4 x 64
                           const float* __restrict__ k,    // 16 x 64
                           const float* __restrict__ wq,   // 64 x 64 [out,in]
                           const float* __restrict__ wk,   // 64 x 64 [out,in]
                           float* __restrict__ attn_out,   // 16 floats (d_out tail)
                           float* __restrict__ weights)    // 16 floats (d_ws)
{
    __shared__ float qpad[16 * 64];
    __shared__ float qh[16 * 64];
    __shared__ float kh[16 * 64];
    __shared__ float attn_s[16];

    const int lane = threadIdx.x;       // 0..31
    const int half = lane >> 4;         // 0 | 1
    const int l16  = lane & 15;

    // Zero-pad q: rows 4..15 must contribute 0 to the WMMA.
    for (int idx = lane; idx < 16 * 64; idx += 32)
        qpad[idx] = (idx < 4 * 64) ? q[idx] : 0.0f;
    __syncthreads();

    // ---- qh = Qpad @ Wq^T ----
    for (int jj = 0; jj < 64; jj += 16) {
        v8f c = {};
        #pragma unroll
        for (int kk = 0; kk < 64; kk += 4) {
            const int kb = kk + 2 * half;
            v2f a;  a.x  = qpad[l16 * 64 + kb];     a.y  = qpad[l16 * 64 + kb + 1];
            const int j  = jj + l16;                 // B[d][j] = Wq[j][d]
            v2f bb; bb.x = wq[j * 64 + kb];         bb.y = wq[j * 64 + kb + 1];
            c = wmma_f32_k4(a, bb, c);
        }
        #pragma unroll
        for (int p = 0; p < 8; ++p)
            qh[(p + 8 * half) * 64 + jj + l16] = c[p];
    }

    // ---- kh = K @ Wk^T ----
    for (int jj = 0; jj < 64; jj += 16) {
        v8f c = {};
        #pragma unroll
        for (int kk = 0; kk < 64; kk += 4) {
            const int kb = kk + 2 * half;
            v2f a;  a.x  = k[l16 * 64 + kb];        a.y  = k[l16 * 64 + kb + 1];
            const int j  = jj + l16;
            v2f bb; bb.x = wk[j * 64 + kb];         bb.y = wk[j * 64 + kb + 1];
            c = wmma_f32_k4(a, bb, c);
        }
        #pragma unroll
        for (int p = 0; p < 8; ++p)
            kh[(p + 8 * half) * 64 + jj + l16] = c[p];
    }
    __syncthreads();

    // ---- scores: attn[r=4b+n] = (qh[b] . kh[r]) / sqrt(64) ----
    if (lane < 16) {
        const int b = lane >> 2;
        float s = 0.0f;
        #pragma unroll
        for (int d = 0; d < 64; ++d)
            s = fmaf(qh[b * 64 + d], kh[lane * 64 + d], s);
        s *= 0.125f;                    // 1/sqrt(D_K), D_K = 64
        attn_s[lane]   = s;
        attn_out[lane] = s;             // attn_flat output (pre-softmax)
    }
    __syncthreads();

    // ---- softmax over n (4 values per batch) ----
    if (lane < 4) {
        const float x0 = attn_s[lane * 4 + 0];
        const float x1 = attn_s[lane * 4 + 1];
        const float x2 = attn_s[lane * 4 + 2];
        const float x3 = attn_s[lane * 4 + 3];
        const float m  = fmaxf(fmaxf(x0, x1), fmaxf(x2, x3));
        const float e0 = expf(x0 - m);
        const float e1 = expf(x1 - m);
        const float e2 = expf(x2 - m);
        const float e3 = expf(x3 - m);
        const float inv = 1.0f / (e0 + e1 + e2 + e3);
        weights[lane * 4 + 0] = e0 * inv;
        weights[lane * 4 + 1] = e1 * inv;
        weights[lane * 4 + 2] = e2 * inv;
        weights[lane * 4 + 3] = e3 * inv;
    }
}

// ---------------------------------------------------------------------------
// Phase 2: res[b, i] = sum_n w[b][n] * v[b, n, i],  i over 64*256*256 floats.
// One float4 per thread: 4 NT B128 loads + 1 NT B128 store. Pure streaming;
// target is the 23.3 TB/s HBM roofline.
// Grid: (4096, 4), block 256  ->  4096*256*4 floats per batch = 4,194,304. ok.
// ---------------------------------------------------------------------------
__global__ __launch_bounds__(256)
void CDAM_combine_kernel(const float* __restrict__ v,    // [4,4,4194304]
                         const float* __restrict__ w,    // [4,4] softmax weights
                         float* __restrict__ out)        // [4,4194304]
{
    const long long SLICE = 64LL * 256 * 256;            // 4,194,304 floats
    const int       b     = blockIdx.y;
    const long long i     = (((long long)blockIdx.x * blockDim.x) + threadIdx.x) * 4;

    // Uniform per-block address -> scalarized by compiler; L2-resident.
    const v4f wv = *(const v4f*)(w + 4 * b);

    const float* vb = v + (long long)b * 4 * SLICE;
    const v4f v0 = __builtin_nontemporal_load((const v4f*)(vb + 0 * SLICE + i));
    const v4f v1 = __builtin_nontemporal_load((const v4f*)(vb + 1 * SLICE + i));
    const v4f v2 = __builtin_nontemporal_load((const v4f*)(vb + 2 * SLICE + i));
    const v4f v3 = __builtin_nontemporal_load((const v4f*)(vb + 3 * SLICE + i));

    v4f r = wv.x * v0;
    r = r + wv.y * v1;
    r = r + wv.z * v2;
    r = r + wv.w * v3;

    __builtin_nontemporal_store(r, (v4f*)(out + (long long)b * SLICE + i));
}

extern "C" void kernel_launch(void* const* d_in, const int* in_sizes, int n_in,
                              void* d_out, int out_size, void* d_ws, size_t ws_size,
                              hipStream_t stream) {
    const float* q  = (const float*)d_in[0];   // (4, 1, 64)
    const float* k  = (const float*)d_in[1];   // (4, 4, 64)
    const float* v  = (const float*)d_in[2];   // (4, 4, 64, 256, 256)
    const float* wq = (const float*)d_in[3];   // (64, 64)
    const float* wk = (const float*)d_in[4];   // (64, 64)

    float* out = (float*)d_out;                // res (16,777,216) ++ attn_flat (16)
    float* ws  = (float*)d_ws;                 // 16 softmax weights

    const long long RES_ELEMS = 4LL * 64 * 256 * 256;   // 16,777,216

    CDAM_attn_wmma_kernel<<<1, 32, 0, stream>>>(q, k, wq, wk, out + RES_ELEMS, ws);

    dim3 grid(4096, 4);
    dim3 block(256);
    CDAM_combine_kernel<<<grid, block, 0, stream>>>(v, ws, out);
}